// SinkhornDistance_43224550867322
// MI455X (gfx1250) — compile-verified
//
#include <hip/hip_runtime.h>
#include <hip/hip_bf16.h>
#include <math.h>

// ---------------------------------------------------------------------------
// Sinkhorn distance, B=8192, D=16, eps=1, 50 iters, thr=1e-3.
// Strategy (MI455X): build K = exp(-C) once with V_WMMA_F32_16X16X4_F32
// (the only matrix op in the problem), store K in d_out's P region (256 MB),
// then stream K twice per iteration (bandwidth-bound, ~256 MB/pass; 192 MB L2
// keeps most of K resident). Finalize P in place and recover C = -ln K
// (exact range: C in [0,16] => K in [e^-16, 1], no underflow).
// ---------------------------------------------------------------------------

typedef float v2f __attribute__((ext_vector_type(2)));
typedef float v8f __attribute__((ext_vector_type(8)));

#define NB 8192
#define DD 16
#define EPS 1.0f
#define ITERS 50
#define THRESH 1e-3f
#define INV_N (1.0f / 8192.0f)

// workspace layout (float offsets); total ~2.3 MB
#define WS_X2   0
#define WS_Y2   8192
#define WS_U    16384
#define WS_UN   24576
#define WS_V    32768
#define WS_WP   40960      // 1024 W block-partials
#define WS_FLAG 41984      // [0]=done (int), [1]=maxdiff bits (int)
#define WS_COLP 49152      // COL_CHUNKS * NB column partials
#define COL_CHUNKS 64
#define ROWS_PER_CHUNK (NB / COL_CHUNKS)   // 128

// ---- setup: row norms, u=v=1, flags=0 -------------------------------------
__global__ void k_setup(const float* __restrict__ x, const float* __restrict__ y,
                        float* __restrict__ ws) {
    const int i = blockIdx.x * blockDim.x + threadIdx.x;   // 0..8191
    float sx = 0.f, sy = 0.f;
#pragma unroll
    for (int k = 0; k < DD; ++k) {
        const float a = x[i * DD + k]; sx = fmaf(a, a, sx);
        const float b = y[i * DD + k]; sy = fmaf(b, b, sy);
    }
    ws[WS_X2 + i] = sx;
    ws[WS_Y2 + i] = sy;
    ws[WS_U + i] = 1.0f;
    ws[WS_V + i] = 1.0f;
    if (i == 0) {
        ((int*)ws)[WS_FLAG + 0] = 0;   // done = false
        ((int*)ws)[WS_FLAG + 1] = 0;   // maxdiff = 0.0f
    }
}

// ---- build K: one wave per 16x16 tile, 4x V_WMMA_F32_16X16X4_F32 ----------
// A (16x4 slice of x): lane m=l&15, hi=l>>4 -> float2 at x[(i0+m)*16 + k0+2*hi]
// B (4x16 slice of y^T): B[k][n]=y[n][k]   -> float2 at y[(j0+n)*16 + k0+2*hi]
// C/D: acc[r] holds element (m = r + 8*hi, n = l&15).
__global__ void k_buildK(const float* __restrict__ x, const float* __restrict__ y,
                         const float* __restrict__ ws, float* __restrict__ Kmat) {
    const int lane = threadIdx.x & 31;
    const int wave = threadIdx.x >> 5;                     // 8 waves / block
    const int j0 = (blockIdx.x * 8 + wave) * 16;
    const int i0 = blockIdx.y * 16;
    const int m  = lane & 15;
    const int hi = lane >> 4;

    v8f acc = {};
#pragma unroll
    for (int kk = 0; kk < 4; ++kk) {
        const int k0 = kk * 4 + 2 * hi;
        v2f a = *(const v2f*)(x + (i0 + m) * DD + k0);
        v2f b = *(const v2f*)(y + (j0 + m) * DD + k0);
        acc = __builtin_amdgcn_wmma_f32_16x16x4_f32(
            /*neg_a=*/false, a, /*neg_b=*/false, b,
            /*c_mod=*/(short)0, acc, /*reuse_a=*/false, /*reuse_b=*/false);
    }

    const float y2j = ws[WS_Y2 + j0 + m];
#pragma unroll
    for (int r = 0; r < 8; ++r) {
        const int i = i0 + r + 8 * hi;
        float c = ws[WS_X2 + i] + y2j - 2.0f * acc[r];
        c = fmaxf(c, 0.0f);
        Kmat[(size_t)i * NB + (j0 + m)] = __expf(-c * (1.0f / EPS));
    }
}

// ---- column pass: partial K^T u over a 128-row chunk ----------------------
__global__ void k_colpass(const float* __restrict__ Kmat, float* __restrict__ ws) {
    const int j = blockIdx.x * 256 + threadIdx.x;
    const int ibase = blockIdx.y * ROWS_PER_CHUNK;
    const float* __restrict__ u = ws + WS_U;
    float s = 0.f;
    for (int ii = 0; ii < ROWS_PER_CHUNK; ++ii) {
        const int i = ibase + ii;
        s = fmaf(Kmat[(size_t)i * NB + j], u[i], s);       // coalesced over j
    }
    ws[WS_COLP + (size_t)blockIdx.y * NB + j] = s;
}

// ---- v update: reduce chunk partials; freeze when done (carry-in) ---------
__global__ void k_vfinal(float* __restrict__ ws) {
    const int j = blockIdx.x * 256 + threadIdx.x;
    float s = 0.f;
#pragma unroll 8
    for (int c = 0; c < COL_CHUNKS; ++c) s += ws[WS_COLP + (size_t)c * NB + j];
    const int done = ((const int*)ws)[WS_FLAG];
    if (!done) ws[WS_V + j] = INV_N / (s + 1e-8f);
}

// ---- row pass: u_new = mu / (K v + 1e-8), wave-per-row; maxdiff -----------
__global__ void k_rowpass(const float* __restrict__ Kmat, float* __restrict__ ws) {
    const int lane = threadIdx.x & 31;
    const int wave = threadIdx.x >> 5;
    const int i = blockIdx.x * 8 + wave;
    const float* __restrict__ v = ws + WS_V;
    const float* __restrict__ Krow = Kmat + (size_t)i * NB;
    float s = 0.f;
    for (int j = lane; j < NB; j += 32) s = fmaf(Krow[j], v[j], s);
#pragma unroll
    for (int off = 16; off > 0; off >>= 1) s += __shfl_xor(s, off, 32);
    if (lane == 0) {
        const float un = INV_N / (s + 1e-8f);
        ws[WS_UN + i] = un;
        const float d = fabsf(un - ws[WS_U + i]);          // >= 0: int max == float max
        atomicMax((int*)ws + WS_FLAG + 1, __float_as_int(d));
    }
}

// ---- u update: freeze when done | conv ------------------------------------
__global__ void k_ufinal(float* __restrict__ ws) {
    const int i = blockIdx.x * 256 + threadIdx.x;
    const int done = ((const int*)ws)[WS_FLAG];
    const float md = __int_as_float(((const int*)ws)[WS_FLAG + 1]);
    if (!(done || (md < THRESH))) ws[WS_U + i] = ws[WS_UN + i];
}

// ---- carry flags into next iteration --------------------------------------
__global__ void k_flags(float* __restrict__ ws) {
    int* f = (int*)ws + WS_FLAG;
    if (__int_as_float(f[1]) < THRESH) f[0] = 1;           // done |= conv
    f[1] = 0;                                              // reset maxdiff
}

// ---- finalize: P = u*K*v in place; W partials with C = -eps*ln(K) ---------
__global__ void k_final(float* __restrict__ Kmat, float* __restrict__ ws) {
    __shared__ float red[8];
    const int lane = threadIdx.x & 31;
    const int wave = threadIdx.x >> 5;
    const int i = blockIdx.x * 8 + wave;
    const float ui = ws[WS_U + i];
    const float* __restrict__ v = ws + WS_V;
    float* __restrict__ row = Kmat + (size_t)i * NB;
    float w = 0.f;
    for (int j = lane; j < NB; j += 32) {
        const float kv = row[j];
        const float p = ui * kv * v[j];
        row[j] = p;                                        // K -> P in place
        const float c = fmaxf(-__logf(kv), 0.0f) * EPS;    // C recovered exactly
        w = fmaf(p, c, w);
    }
#pragma unroll
    for (int off = 16; off > 0; off >>= 1) w += __shfl_xor(w, off, 32);
    if (lane == 0) red[wave] = w;
    __syncthreads();
    if (threadIdx.x == 0) {
        float t = 0.f;
#pragma unroll
        for (int q = 0; q < 8; ++q) t += red[q];
        ws[WS_WP + blockIdx.x] = t;
    }
}

// ---- deterministic reduction of 1024 W partials ---------------------------
__global__ void k_wreduce(const float* __restrict__ ws, float* __restrict__ out) {
    __shared__ float red[32];
    const int t = threadIdx.x;                             // 1024 threads
    float s = ws[WS_WP + t];
#pragma unroll
    for (int off = 16; off > 0; off >>= 1) s += __shfl_xor(s, off, 32);
    if ((t & 31) == 0) red[t >> 5] = s;
    __syncthreads();
    if (t < 32) {
        float z = red[t];
#pragma unroll
        for (int off = 16; off > 0; off >>= 1) z += __shfl_xor(z, off, 32);
        if (t == 0) out[0] = z;
    }
}

extern "C" void kernel_launch(void* const* d_in, const int* in_sizes, int n_in,
                              void* d_out, int out_size, void* d_ws, size_t ws_size,
                              hipStream_t stream) {
    (void)in_sizes; (void)n_in; (void)out_size; (void)ws_size;
    const float* x = (const float*)d_in[0];
    const float* y = (const float*)d_in[1];
    float* out  = (float*)d_out;
    float* Kmat = out + 1;            // P region doubles as K storage
    float* ws   = (float*)d_ws;

    k_setup<<<NB / 256, 256, 0, stream>>>(x, y, ws);
    k_buildK<<<dim3(NB / (16 * 8), NB / 16), 256, 0, stream>>>(x, y, ws, Kmat);

    for (int it = 0; it < ITERS; ++it) {
        k_colpass<<<dim3(NB / 256, COL_CHUNKS), 256, 0, stream>>>(Kmat, ws);
        k_vfinal<<<NB / 256, 256, 0, stream>>>(ws);
        k_rowpass<<<NB / 8, 256, 0, stream>>>(Kmat, ws);
        k_ufinal<<<NB / 256, 256, 0, stream>>>(ws);
        k_flags<<<1, 1, 0, stream>>>(ws);
    }

    k_final<<<NB / 8, 256, 0, stream>>>(Kmat, ws);
    k_wreduce<<<1, 1024, 0, stream>>>(ws, out);
}